// KalmanFilter_46445776339201
// MI455X (gfx1250) — compile-verified
//
#include <hip/hip_runtime.h>

typedef float v2f __attribute__((ext_vector_type(2)));
typedef float v8f __attribute__((ext_vector_type(8)));

constexpr int BS = 128, NV = 32, T = 128, SD = 64;
constexpr int NTHREADS = 256;   // 8 waves of 32 (wave32)
constexpr int NWAVES = 8;

// C[M x N] = A[M x K] * (TRANSB ? B^T : B) (+ Cin), all fp32 via V_WMMA_F32_16X16X4_F32.
// Output 16x16 tiles distributed across the 8 waves; branches are wave-uniform so EXEC
// stays all-ones inside the WMMA region.
template<int M, int N, int K, bool TRANSB>
__device__ __forceinline__ void mm_f32(const float* __restrict__ A, int lda,
                                       const float* __restrict__ B, int ldb,
                                       const float* __restrict__ Cin, int ldcin,
                                       float* __restrict__ Cout, int ldcout,
                                       int wave, int lane)
{
    const int m16  = lane & 15;   // row (A) / col (B,C) within tile
    const int half = lane >> 4;   // selects K-pair (A/B) or row-block (C)
    constexpr int tilesN = N / 16;
    constexpr int tiles  = (M / 16) * tilesN;
    for (int tt = wave; tt < tiles; tt += NWAVES) {
        const int ti = (tt / tilesN) * 16;
        const int tj = (tt % tilesN) * 16;
        v8f c = {};
        if (Cin) {
            #pragma unroll
            for (int r = 0; r < 8; ++r)
                c[r] = Cin[(size_t)(ti + r + 8 * half) * ldcin + (tj + m16)];
        }
        #pragma unroll
        for (int k0 = 0; k0 < K; k0 += 4) {
            v2f a, b;
            // A 16x4 tile: lane m16 holds row, half selects K pair {k0,k0+1} / {k0+2,k0+3}
            a.x = A[(ti + m16) * lda + k0 + 2 * half];
            a.y = A[(ti + m16) * lda + k0 + 2 * half + 1];
            if (TRANSB) {   // B^T[k][n] = B[n][k]
                b.x = B[(tj + m16) * ldb + k0 + 2 * half];
                b.y = B[(tj + m16) * ldb + k0 + 2 * half + 1];
            } else {        // B 4x16 tile: lane m16 holds col, half selects K pair
                b.x = B[(k0 + 2 * half) * ldb + tj + m16];
                b.y = B[(k0 + 2 * half + 1) * ldb + tj + m16];
            }
            c = __builtin_amdgcn_wmma_f32_16x16x4_f32(false, a, false, b,
                                                      (short)0, c, false, false);
        }
        #pragma unroll
        for (int r = 0; r < 8; ++r)
            Cout[(ti + r + 8 * half) * ldcout + (tj + m16)] = c[r];
    }
}

__global__ __launch_bounds__(NTHREADS)
void kf_kernel(const float* __restrict__ x, const float* __restrict__ gF,
               const float* __restrict__ gH, const float* __restrict__ gR,
               const float* __restrict__ gQ, float* __restrict__ out)
{
    __shared__ float sF[SD * SD];        // F          64x64
    __shared__ float sP[SD * SD];        // P          64x64 (state)
    __shared__ float sT1[SD * SD];       // K@H -> p1  64x64
    __shared__ float sT2[SD * SD];       // temp       64x64
    __shared__ float sH[NV * SD];        // H          32x64
    __shared__ float sK[SD * NV];        // K          64x32
    __shared__ float sPHt[SD * NV];      // P@H^T, then K@R   64x32
    __shared__ float sR[NV * NV];        // R          32x32
    __shared__ float sAug[NV * 2 * NV];  // [S | I] -> [_ | S^-1]  32x64
    __shared__ float sm[SD], smu[SD], sresid[NV];

    const int tid  = threadIdx.x;
    const int wave = tid >> 5;
    const int lane = tid & 31;
    const int b    = blockIdx.x;

    // ---- init: constants to LDS, P = I, m = 0, out[:, :, 0] = 0 ----
    for (int i = tid; i < SD * SD; i += NTHREADS) {
        sF[i] = gF[i];
        sP[i] = ((i >> 6) == (i & 63)) ? 1.0f : 0.0f;
    }
    for (int i = tid; i < NV * SD; i += NTHREADS) sH[i] = gH[i];
    for (int i = tid; i < NV * NV; i += NTHREADS) sR[i] = gR[i];
    if (tid < SD) sm[tid] = 0.0f;
    if (tid < NV) out[((size_t)b * NV + tid) * T + 0] = 0.0f;
    __syncthreads();

    #pragma unroll 1
    for (int t = 0; t < T - 1; ++t) {
        // 1) PHt = P @ H^T                              [64x32]
        mm_f32<SD, NV, SD, true>(sP, SD, sH, SD, nullptr, 0, sPHt, NV, wave, lane);
        __syncthreads();

        // 2) Aug[:, :32] = H @ PHt + R;  Aug[:, 32:] = I;  resid = o - H m
        mm_f32<NV, NV, SD, false>(sH, SD, sPHt, NV, sR, NV, sAug, 2 * NV, wave, lane);
        for (int i2 = tid; i2 < NV * NV; i2 += NTHREADS) {
            int ri = i2 >> 5, rj = i2 & 31;
            sAug[ri * 64 + 32 + rj] = (ri == rj) ? 1.0f : 0.0f;
        }
        if (tid >= 128 && tid < 160) {          // wave 4: residual
            int v = tid - 128;
            float acc = 0.0f;
            #pragma unroll
            for (int k2 = 0; k2 < SD; ++k2) acc += sH[v * SD + k2] * sm[k2];
            sresid[v] = x[((size_t)b * NV + v) * T + t] - acc;
        }
        __syncthreads();

        // 3) Gauss-Jordan (no pivoting; S is SPD): Aug[:, 32:] = S^-1
        {
            const int gi  = tid >> 3;           // row 0..31
            const int gj0 = (tid & 7) << 3;     // 8 columns per thread
            for (int k = 0; k < NV; ++k) {
                float pivinv = 1.0f / sAug[k * 64 + k];
                float factor = sAug[gi * 64 + k] * pivinv;
                __syncthreads();
                if (gi != k) {
                    #pragma unroll
                    for (int jj = 0; jj < 8; ++jj)
                        sAug[gi * 64 + gj0 + jj] -= factor * sAug[k * 64 + gj0 + jj];
                }
                __syncthreads();
                if (gi == k) {
                    #pragma unroll
                    for (int jj = 0; jj < 8; ++jj)
                        sAug[k * 64 + gj0 + jj] *= pivinv;
                }
                __syncthreads();
            }
        }

        // 4) K = PHt @ S^-1                             [64x32]
        mm_f32<SD, NV, NV, false>(sPHt, NV, sAug + 32, 2 * NV, nullptr, 0, sK, NV, wave, lane);
        __syncthreads();

        // 5) T1 = K @ H [64x64];  m_u = m + K resid
        mm_f32<SD, SD, NV, false>(sK, NV, sH, SD, nullptr, 0, sT1, SD, wave, lane);
        if (tid < SD) {
            float acc = sm[tid];
            #pragma unroll
            for (int v = 0; v < NV; ++v) acc += sK[tid * NV + v] * sresid[v];
            smu[tid] = acc;
        }
        __syncthreads();

        // 6) p1 = I - T1 (in place);  KR = K @ R -> sPHt (PHt dead)
        for (int i2 = tid; i2 < SD * SD; i2 += NTHREADS) {
            float v = sT1[i2];
            sT1[i2] = (((i2 >> 6) == (i2 & 63)) ? 1.0f : 0.0f) - v;
        }
        mm_f32<SD, NV, NV, false>(sK, NV, sR, NV, nullptr, 0, sPHt, NV, wave, lane);
        __syncthreads();

        // 7) T2 = p1 @ P
        mm_f32<SD, SD, SD, false>(sT1, SD, sP, SD, nullptr, 0, sT2, SD, wave, lane);
        __syncthreads();

        // 8) P_u = T2 @ p1^T; 9) P_u += KR @ K^T
        //    (identical per-wave tile ownership in both calls -> no barrier needed between)
        mm_f32<SD, SD, SD, true>(sT2, SD, sT1, SD, nullptr, 0, sP, SD, wave, lane);
        mm_f32<SD, SD, NV, true>(sPHt, NV, sK, NV, sP, SD, sP, SD, wave, lane);
        __syncthreads();

        // 10) T2 = F @ P_u;  m_p = F m_u -> sm
        mm_f32<SD, SD, SD, false>(sF, SD, sP, SD, nullptr, 0, sT2, SD, wave, lane);
        if (tid < SD) {
            float acc = 0.0f;
            #pragma unroll
            for (int k2 = 0; k2 < SD; ++k2) acc += sF[tid * SD + k2] * smu[k2];
            sm[tid] = acc;
        }
        __syncthreads();

        // 11) P = T2 @ F^T + Q (Q streamed from global as C-init);  y = H m_p
        mm_f32<SD, SD, SD, true>(sT2, SD, sF, SD, gQ, SD, sP, SD, wave, lane);
        if (tid < NV) {
            float acc = 0.0f;
            #pragma unroll
            for (int k2 = 0; k2 < SD; ++k2) acc += sH[tid * SD + k2] * sm[k2];
            out[((size_t)b * NV + tid) * T + (t + 1)] = acc;
        }
        __syncthreads();
    }
}

extern "C" void kernel_launch(void* const* d_in, const int* in_sizes, int n_in,
                              void* d_out, int out_size, void* d_ws, size_t ws_size,
                              hipStream_t stream) {
    (void)in_sizes; (void)n_in; (void)out_size; (void)d_ws; (void)ws_size;
    const float* x = (const float*)d_in[0];
    const float* F = (const float*)d_in[1];
    const float* H = (const float*)d_in[2];
    const float* R = (const float*)d_in[3];
    const float* Q = (const float*)d_in[4];
    float* out = (float*)d_out;
    kf_kernel<<<dim3(BS), dim3(NTHREADS), 0, stream>>>(x, F, H, R, Q, out);
}